// SortPool2D_75608604279123
// MI455X (gfx1250) — compile-verified
//
#include <hip/hip_runtime.h>
#include <stdint.h>

typedef float v4f __attribute__((ext_vector_type(4)));

#define NB 16
#define NH 224
#define NW 224
#define NC 256
#define OHH 112
#define OWW 112
#define NC4 64                         // NC / 4 channel-groups (float4)
#define THREADS 256                    // 8 wave32s
#define ITERS ((OWW * NC4) / THREADS)  // 28 iterations per output row

// ---------------------------------------------------------------------------
// Tiny kernel: softmax over the 4 pool weights -> d_ws[0..3]
// ---------------------------------------------------------------------------
__global__ void softmax4_kernel(const float* __restrict__ pw, float* __restrict__ w) {
    if (threadIdx.x == 0) {
        float p0 = pw[0], p1 = pw[1], p2 = pw[2], p3 = pw[3];
        float m  = fmaxf(fmaxf(p0, p1), fmaxf(p2, p3));
        float e0 = expf(p0 - m), e1 = expf(p1 - m), e2 = expf(p2 - m), e3 = expf(p3 - m);
        float inv = 1.0f / (e0 + e1 + e2 + e3);
        w[0] = e0 * inv; w[1] = e1 * inv; w[2] = e2 * inv; w[3] = e3 * inv;
    }
}

// CDNA5 async data mover: global -> LDS, 16 bytes per lane, tracked by ASYNCcnt.
// GVS mode: 64-bit SGPR base + signed 32-bit per-lane byte offset.
// Input is a one-pass stream -> non-temporal hint keeps L2 clean.
__device__ __forceinline__ void async_load16(uint32_t lds_addr, uint32_t goff,
                                             const float* base) {
    asm volatile("global_load_async_to_lds_b128 %0, %1, %2 th:TH_LOAD_NT"
                 :
                 : "v"(lds_addr), "v"(goff), "s"(base)
                 : "memory");
}

// ---------------------------------------------------------------------------
// Main kernel: 2x2 sort-pool, async-LDS double-buffered streaming pipeline.
// One block handles one (batch, output-row) pair: 112 x 64 float4 outputs.
// ---------------------------------------------------------------------------
__launch_bounds__(THREADS)
__global__ void sortpool_kernel(const float* __restrict__ x,
                                const float* __restrict__ w4,
                                float* __restrict__ out) {
    // [double-buffer][quadrant TL/TR/BL/BR][THREADS float4]  = 32 KB LDS
    __shared__ __align__(16) float stage[2][4][THREADS * 4];

    const uint32_t tid = threadIdx.x;
    const uint32_t b   = blockIdx.x / OHH;
    const uint32_t oh  = blockIdx.x % OHH;
    const uint32_t ih  = oh * 2u;

    const v4f wv = *reinterpret_cast<const v4f*>(w4);  // softmax'd weights (ascending order)

    const uint32_t rowB    = (uint32_t)NW * NC * 4u;   // bytes per input image row
    const uint32_t pixB    = (uint32_t)NC * 4u;        // bytes per input pixel
    const uint32_t rowBase = (b * NH + ih) * rowB;     // byte offset of input row ih

    // Per-lane LDS byte addresses for the 8 staging slots (flat-ptr low 32 bits
    // == DS byte address per the gfx1250 LDS aperture mapping).
    uint32_t laddr[2][4];
#pragma unroll
    for (int bb = 0; bb < 2; ++bb)
#pragma unroll
        for (int k = 0; k < 4; ++k)
            laddr[bb][k] = (uint32_t)(uintptr_t)(&stage[bb][k][tid * 4u]);

    // Issue iteration i's four async loads into buffer `buf`.
    auto issue = [&](uint32_t i, uint32_t buf) {
        const uint32_t li = i * THREADS + tid;
        const uint32_t ow = li >> 6;          // output column
        const uint32_t c4 = li & 63u;         // float4 channel group
        const uint32_t o0 = rowBase + ow * (2u * pixB) + c4 * 16u;  // top-left
        async_load16(laddr[buf][0], o0,               x);           // TL
        async_load16(laddr[buf][1], o0 + pixB,        x);           // TR
        async_load16(laddr[buf][2], o0 + rowB,        x);           // BL
        async_load16(laddr[buf][3], o0 + rowB + pixB, x);           // BR
    };

    issue(0u, 0u);

#pragma unroll 1
    for (uint32_t i = 0; i < ITERS; ++i) {
        const uint32_t buf = i & 1u;
        if (i + 1u < ITERS) {
            issue(i + 1u, buf ^ 1u);                       // prefetch next buffer
            // 4 new loads outstanding; async loads complete in order, so
            // ASYNCcnt <= 4 means this iteration's 4 loads have landed in LDS.
            asm volatile("s_wait_asynccnt 4" ::: "memory");
        } else {
            asm volatile("s_wait_asynccnt 0" ::: "memory");
        }

        // ds_load_b128 readback: 16 B lane stride -> conflict-free.
        const v4f A  = reinterpret_cast<const v4f*>(&stage[buf][0][0])[tid];
        const v4f Bv = reinterpret_cast<const v4f*>(&stage[buf][1][0])[tid];
        const v4f Cv = reinterpret_cast<const v4f*>(&stage[buf][2][0])[tid];
        const v4f Dv = reinterpret_cast<const v4f*>(&stage[buf][3][0])[tid];

        // 4-element sorting network (ascending), elementwise over 4 channels.
        const v4f t0 = __builtin_elementwise_min(A, Bv);
        const v4f t1 = __builtin_elementwise_max(A, Bv);
        const v4f t2 = __builtin_elementwise_min(Cv, Dv);
        const v4f t3 = __builtin_elementwise_max(Cv, Dv);
        const v4f q0 = __builtin_elementwise_min(t0, t2);   // smallest
        const v4f m1 = __builtin_elementwise_max(t0, t2);
        const v4f m2 = __builtin_elementwise_min(t1, t3);
        const v4f q3 = __builtin_elementwise_max(t1, t3);   // largest
        const v4f q1 = __builtin_elementwise_min(m1, m2);
        const v4f q2 = __builtin_elementwise_max(m1, m2);

        // Weighted sum with softmax'd weights (sorted ascending matches ref).
        v4f r = q0 * wv.x;
        r += q1 * wv.y;
        r += q2 * wv.z;
        r += q3 * wv.w;

        const uint32_t li = i * THREADS + tid;
        const uint32_t ow = li >> 6;
        const uint32_t c4 = li & 63u;
        float* op = out + (((b * OHH + oh) * OWW + ow) * NC + c4 * 4u);
        // Streamed once: non-temporal b128 store, keep L2 for the input stream.
        __builtin_nontemporal_store(r, reinterpret_cast<v4f*>(op));
    }
}

// ---------------------------------------------------------------------------
// Harness entry point
// ---------------------------------------------------------------------------
extern "C" void kernel_launch(void* const* d_in, const int* in_sizes, int n_in,
                              void* d_out, int out_size, void* d_ws, size_t ws_size,
                              hipStream_t stream) {
    const float* x  = (const float*)d_in[0];   // (16,224,224,256) fp32
    const float* pw = (const float*)d_in[1];   // (1,1,1,1,4) fp32
    float* out = (float*)d_out;                // (16,112,112,256) fp32
    float* w   = (float*)d_ws;                 // 4 softmax'd weights

    softmax4_kernel<<<1, 32, 0, stream>>>(pw, w);
    sortpool_kernel<<<NB * OHH, THREADS, 0, stream>>>(x, w, out);
}